// GAT_6227702579851
// MI455X (gfx1250) — compile-verified
//
#include <hip/hip_runtime.h>

// GAT 3-layer forward for MI455X (gfx1250, wave32, WMMA).
#define NNODES 50000
#define NEDGES 250000
#define ETOT   (NEDGES + NNODES)   // self-loops appended
#define FIN    128
#define HID    64
#define HEADS  8
#define FOUT   3
#define NEG_SLOPE 0.2f

typedef __attribute__((ext_vector_type(2))) float v2f;
typedef __attribute__((ext_vector_type(4))) float v4f;
typedef __attribute__((ext_vector_type(8))) float v8f;

// ---------------------------------------------------------------------------
// C[M,NCOLS] = A[M,K] @ B[K,NCOLS] with fp32 WMMA 16x16x4.
// Compile-time K/NCOLS -> constant-stride address math, no masking, EXEC all-1s.
// Each wave computes MT stacked 16-row tiles in one 16-col stripe, so one B
// fragment feeds MT WMMAs per k-step (5x B-stream reuse; 3125 = 5^5 row tiles).
// ---------------------------------------------------------------------------
template <int K, int NCOLS, int MT>
__global__ void gemm_wmma_f32(const float* __restrict__ A,
                              const float* __restrict__ B,
                              float* __restrict__ C,
                              int mtilesM /* = (M/16)/MT */) {
    constexpr int NTN = NCOLS / 16;
    const int wave = blockIdx.x * (blockDim.x >> 5) + (threadIdx.x >> 5);
    if (wave >= mtilesM * NTN) return;          // whole-wave exit, EXEC stays full

    const int tm0  = (wave / NTN) * MT;         // first row-tile of this wave
    const int tn   = wave % NTN;
    const int lane = threadIdx.x & 31;
    const int l16  = lane & 15;
    const int hi   = lane >> 4;                 // 0: k%4 in {0,1}; 1: k%4 in {2,3}
    const int col  = tn * 16 + l16;

    // A frag (16x4 f32): lane holds A[tm*16+l16][k+2*hi .. +1] as an 8B float2.
    const float* __restrict__ aBase = A + ((size_t)tm0 * 16 + l16) * K + 2 * hi;
    // B frag (4x16 f32): lane holds B[k+2*hi][col], B[k+2*hi+1][col].
    const float* __restrict__ bBase = B + (size_t)(2 * hi) * NCOLS + col;

    v8f acc[MT];
    #pragma unroll
    for (int m = 0; m < MT; ++m) acc[m] = {};

    #pragma unroll 4
    for (int k = 0; k < K; k += 4) {
        v2f b;
        b.x = bBase[(size_t)k * NCOLS];
        b.y = bBase[(size_t)k * NCOLS + NCOLS];
        #pragma unroll
        for (int m = 0; m < MT; ++m) {
            v2f a = *(const v2f*)(aBase + (size_t)m * 16 * K + k);
            acc[m] = __builtin_amdgcn_wmma_f32_16x16x4_f32(
                false, a, false, b, (short)0, acc[m], false, false);
        }
    }

    // C/D 16x16 f32: VGPR r holds M = tm*16 + r + 8*hi, N = col.
    #pragma unroll
    for (int m = 0; m < MT; ++m) {
        float* __restrict__ crow =
            C + ((size_t)(tm0 + m) * 16 + hi * 8) * NCOLS + col;
        #pragma unroll
        for (int r = 0; r < 8; ++r)
            crow[(size_t)r * NCOLS] = acc[m][r];
    }
}

// Tiny tail GEMM (layer 3: 64 -> 3). One thread per output element.
__global__ void gemm_small(const float* __restrict__ A, const float* __restrict__ B,
                           float* __restrict__ C, int K, int NC) {
    const int t = blockIdx.x * blockDim.x + threadIdx.x;   // node*NC + c
    if (t >= NNODES * NC) return;
    const int c = t % NC;
    const float* __restrict__ a = A + (size_t)(t / NC) * K;
    float s = 0.0f;
    for (int k = 0; k < K; ++k) s = fmaf(a[k], B[k * NC + c], s);
    C[t] = s;
}

// Per-(node,head): e_s = <h, a_src>, e_d = <h, a_dst>. H,C compile-time.
template <int H, int C>
__global__ void attn_coef(const float* __restrict__ h,
                          const float* __restrict__ a_src,
                          const float* __restrict__ a_dst,
                          float* __restrict__ es, float* __restrict__ ed) {
    const int t = blockIdx.x * blockDim.x + threadIdx.x;   // t = node*H + head
    if (t >= NNODES * H) return;
    const int hh = t % H;                                  // H pow2 -> mask
    const float* __restrict__ hp = h + (size_t)t * C;      // [n,H,C] contiguous
    const float* __restrict__ as = a_src + hh * C;
    const float* __restrict__ ad = a_dst + hh * C;
    float s = 0.0f, d = 0.0f;
    #pragma unroll
    for (int c = 0; c < C; ++c) {
        const float v = hp[c];
        s = fmaf(v, as[c], s);
        d = fmaf(v, ad[c], d);
    }
    es[t] = s;
    ed[t] = d;
}

// Per-(edge,head): w = exp(leaky_relu(e_s[src]+e_d[dst])); denom[dst] += w.
// Max-shift cancels exactly in alpha = w/denom and |e| is O(1) at these scales.
template <int H>
__global__ void edge_weights(const int* __restrict__ srcI, const int* __restrict__ dstI,
                             const float* __restrict__ es, const float* __restrict__ ed,
                             float* __restrict__ wbuf, float* __restrict__ denom) {
    const int t = blockIdx.x * blockDim.x + threadIdx.x;   // t = edge*H + head
    if (t >= ETOT * H) return;
    const int e = t / H, hh = t % H;                       // H pow2 -> shift/mask
    int s, d;
    if (e < NEDGES) { s = srcI[e]; d = dstI[e]; }
    else            { s = d = e - NEDGES; }                // self-loop
    float x = es[(size_t)s * H + hh] + ed[(size_t)d * H + hh];
    x = (x > 0.0f) ? x : NEG_SLOPE * x;
    const float w = __expf(x);
    wbuf[t] = w;
    atomicAdd(&denom[(size_t)d * H + hh], w);
}

// Per-(edge, VEC-channel chunk): out[dst,h,c..c+VEC) += alpha * h[src,h,c..c+VEC).
// One b128 gather + VEC atomic f32 adds per thread; edge index / alpha / address
// math amortized over VEC channels. Requires C % VEC == 0.
template <int H, int C, int VEC>
__global__ void aggregate(const int* __restrict__ srcI, const int* __restrict__ dstI,
                          const float* __restrict__ hfeat,
                          const float* __restrict__ wbuf, const float* __restrict__ denom,
                          float* __restrict__ out) {
    constexpr int HC = H * C;
    constexpr int CH = HC / VEC;                           // chunks per edge
    const long long t = (long long)blockIdx.x * blockDim.x + threadIdx.x;
    if (t >= (long long)ETOT * CH) return;
    const int e  = (int)(t / CH);                          // pow2 CH -> shift
    const int r0 = ((int)(t % CH)) * VEC;
    const int hh = r0 / C;
    int s, d;
    if (e < NEDGES) { s = srcI[e]; d = dstI[e]; }
    else            { s = d = e - NEDGES; }
    const float alpha = wbuf[(size_t)e * H + hh] / denom[(size_t)d * H + hh];
    const float* __restrict__ hp = hfeat + (size_t)s * HC + r0;
    float* __restrict__ op       = out   + (size_t)d * HC + r0;
    if constexpr (VEC == 4) {
        const v4f hv = *(const v4f*)hp;                    // global_load_b128
        #pragma unroll
        for (int i = 0; i < 4; ++i)
            atomicAdd(op + i, alpha * hv[i]);
    } else {
        #pragma unroll
        for (int i = 0; i < VEC; ++i)
            atomicAdd(op + i, alpha * hp[i]);
    }
}

// out = (relu?)(out + bias); HC compile-time so t%HC folds.
template <int HC>
__global__ void finalize(float* __restrict__ out, const float* __restrict__ bias,
                         int relu) {
    const int t = blockIdx.x * blockDim.x + threadIdx.x;
    if (t >= NNODES * HC) return;
    float v = out[t] + bias[t % HC];
    if (relu) v = fmaxf(v, 0.0f);
    out[t] = v;
}

static inline int cdiv(long long a, int b) { return (int)((a + b - 1) / b); }

extern "C" void kernel_launch(void* const* d_in, const int* in_sizes, int n_in,
                              void* d_out, int out_size, void* d_ws, size_t ws_size,
                              hipStream_t stream) {
    const float* x   = (const float*)d_in[0];
    const int*   ei  = (const int*)  d_in[1];     // [2,E] flat: src row then dst row
    const float* W1  = (const float*)d_in[2];
    const float* as1 = (const float*)d_in[3];
    const float* ad1 = (const float*)d_in[4];
    const float* b1  = (const float*)d_in[5];
    const float* W2  = (const float*)d_in[6];
    const float* as2 = (const float*)d_in[7];
    const float* ad2 = (const float*)d_in[8];
    const float* b2  = (const float*)d_in[9];
    const float* W3  = (const float*)d_in[10];
    const float* as3 = (const float*)d_in[11];
    const float* ad3 = (const float*)d_in[12];
    const float* b3  = (const float*)d_in[13];
    float* out = (float*)d_out;

    const int* srcI = ei;
    const int* dstI = ei + NEDGES;

    // Workspace layout (floats): ~220 MB total
    const size_t BIG = (size_t)NNODES * (HEADS * HID);     // 25.6M
    float* bufA = (float*)d_ws;            // GEMM output h (per layer)
    float* bufB = bufA + BIG;              // aggregation accumulator / next-layer x
    float* es   = bufB + BIG;              // N*HEADS
    float* ed   = es + (size_t)NNODES * HEADS;
    float* wb   = ed + (size_t)NNODES * HEADS;             // ETOT*HEADS
    float* dn   = wb + (size_t)ETOT * HEADS;               // N*HEADS

    const int TB = 256;
    const int WPB = TB / 32;               // waves per block
    constexpr int MT = 5;                  // row-tiles per wave (3125 = 5*625)
    const int mtilesM = (NNODES / 16) / MT;                // 625

    // ---------------- Layer 1: 128 -> 8 x 64 (concat 512) ----------------
    {
        constexpr int H = HEADS, C = HID, HC = H * C;
        const int waves = mtilesM * (HC / 16);
        gemm_wmma_f32<FIN, HC, MT>
            <<<cdiv(waves, WPB), TB, 0, stream>>>(x, W1, bufA, mtilesM);
        attn_coef<H, C><<<cdiv((long long)NNODES * H, TB), TB, 0, stream>>>(
            bufA, as1, ad1, es, ed);
        hipMemsetAsync(dn, 0, (size_t)NNODES * H * sizeof(float), stream);
        hipMemsetAsync(bufB, 0, (size_t)NNODES * HC * sizeof(float), stream);
        edge_weights<H><<<cdiv((long long)ETOT * H, TB), TB, 0, stream>>>(
            srcI, dstI, es, ed, wb, dn);
        aggregate<H, C, 4><<<cdiv((long long)ETOT * (HC / 4), TB), TB, 0, stream>>>(
            srcI, dstI, bufA, wb, dn, bufB);
        finalize<HC><<<cdiv((long long)NNODES * HC, TB), TB, 0, stream>>>(
            bufB, b1, 1);                                  // x1 = relu(...)
    }

    // ---------------- Layer 2: 512 -> 64, 1 head ----------------
    {
        constexpr int H = 1, C = HID, HC = H * C;
        const int waves = mtilesM * (HC / 16);
        gemm_wmma_f32<HEADS * HID, HC, MT>
            <<<cdiv(waves, WPB), TB, 0, stream>>>(bufB, W2, bufA, mtilesM);
        attn_coef<H, C><<<cdiv((long long)NNODES * H, TB), TB, 0, stream>>>(
            bufA, as2, ad2, es, ed);
        hipMemsetAsync(dn, 0, (size_t)NNODES * H * sizeof(float), stream);
        hipMemsetAsync(bufB, 0, (size_t)NNODES * HC * sizeof(float), stream);
        edge_weights<H><<<cdiv((long long)ETOT * H, TB), TB, 0, stream>>>(
            srcI, dstI, es, ed, wb, dn);
        aggregate<H, C, 4><<<cdiv((long long)ETOT * (HC / 4), TB), TB, 0, stream>>>(
            srcI, dstI, bufA, wb, dn, bufB);
        finalize<HC><<<cdiv((long long)NNODES * HC, TB), TB, 0, stream>>>(
            bufB, b2, 1);                                  // x2 = relu(...)
    }

    // ---------------- Layer 3: 64 -> 3, 1 head ----------------
    {
        constexpr int H = 1, C = FOUT, HC = H * C;
        gemm_small<<<cdiv((long long)NNODES * HC, TB), TB, 0, stream>>>(
            bufB, W3, bufA, HID, HC);
        attn_coef<H, C><<<cdiv((long long)NNODES * H, TB), TB, 0, stream>>>(
            bufA, as3, ad3, es, ed);
        hipMemsetAsync(dn, 0, (size_t)NNODES * H * sizeof(float), stream);
        hipMemsetAsync(out, 0, (size_t)NNODES * HC * sizeof(float), stream);
        edge_weights<H><<<cdiv((long long)ETOT * H, TB), TB, 0, stream>>>(
            srcI, dstI, es, ed, wb, dn);
        aggregate<H, C, 1><<<cdiv((long long)ETOT * HC, TB), TB, 0, stream>>>(
            srcI, dstI, bufA, wb, dn, out);
        finalize<HC><<<cdiv((long long)NNODES * HC, TB), TB, 0, stream>>>(
            out, b3, 0);                                   // no relu on output
    }
}